// ConditionMask_19198503813720
// MI455X (gfx1250) — compile-verified
//
#include <hip/hip_runtime.h>

typedef __attribute__((ext_vector_type(2))) float v2f;
typedef __attribute__((ext_vector_type(8))) float v8f;

#define EPS 1e-8f

// Problem constants
// B=16, K=64, L=512, TIME_DIM=128, FEAT_DIM=16, out = [16, 32, 64, 512] f32
//
// Workspace layout (floats):
//   [0..127]            wt      (folded time weights)
//   [128..191]          wf      (folded feature weights)
//   [192..194]          ws_stat (folded stat weights)
//   [195]               C       (folded bias constant)
//   [256   .. 256+8191] m       (per-(b,l) column mean over K)
//   [8448  .. 16639]    val_p   (partial val: T + F + ws0*mean + ws1*std + C)
//   [16640 .. 24831]    val     (final per-(b,l) scalar)

// ---------------------------------------------------------------------------
// Fold W_final into the three projections (tiny, one block).
// ---------------------------------------------------------------------------
__global__ __launch_bounds__(256) void prep_kernel(
    const float* __restrict__ W_stat, const float* __restrict__ b_stat,
    const float* __restrict__ W_time, const float* __restrict__ b_time,
    const float* __restrict__ W_feat, const float* __restrict__ b_feat,
    const float* __restrict__ W_final, const float* __restrict__ b_final,
    float* __restrict__ w) {
  int t = threadIdx.x;
  if (t < 128) {
    // wt[i] = sum_o W_time[o,i] * Wf[o]
    float s = 0.f;
    for (int o = 0; o < 128; ++o) s += W_time[o * 128 + t] * W_final[o];
    w[t] = s;
  } else if (t < 192) {
    // wf[k] = sum_f W_feat[f,k] * Wf[128+f]
    int k = t - 128;
    float s = 0.f;
    for (int f = 0; f < 16; ++f) s += W_feat[f * 64 + k] * W_final[128 + f];
    w[128 + k] = s;
  } else if (t < 195) {
    // ws[s] = sum_o W_stat[o,s] * Wf[144+o]
    int sidx = t - 192;
    float s = 0.f;
    for (int o = 0; o < 128; ++o) s += W_stat[o * 3 + sidx] * W_final[144 + o];
    w[192 + sidx] = s;
  } else if (t == 195) {
    // C = b_final + b_time.Wf_t + b_feat.Wf_f + b_stat.Wf_s
    float s = b_final[0];
    for (int o = 0; o < 128; ++o) s += b_time[o] * W_final[o];
    for (int f = 0; f < 16; ++f) s += b_feat[f] * W_final[128 + f];
    for (int o = 0; o < 128; ++o) s += b_stat[o] * W_final[144 + o];
    w[195] = s;
  }
}

// ---------------------------------------------------------------------------
// Per-(b,l) K-reductions (mean/std/colmean/feature-dot) + WMMA f32 16x16x4
// for the 128-d time_embed projection. 32 blocks x 256 threads; each thread
// owns one (b,l); each wave's WMMA covers its 32 l-values via two chains.
// ---------------------------------------------------------------------------
__global__ __launch_bounds__(256) void stats_wmma_kernel(
    const float* __restrict__ obs, const float* __restrict__ tmask,
    const float* __restrict__ te, const float* __restrict__ fe,
    const float* __restrict__ w, float* __restrict__ m_out,
    float* __restrict__ valp_out) {
  const int b    = blockIdx.x >> 1;
  const int l0   = (blockIdx.x & 1) * 256;
  const int lane = threadIdx.x & 31;
  const int wave = threadIdx.x >> 5;
  const int l    = l0 + wave * 32 + lane;

  // ---- scalar K-reductions: lane-contiguous along l => coalesced ----
  const size_t base = (size_t)b * 64 * 512 + l;
  const float* op = obs + base;
  const float* tp = tmask + base;
  const float* fp = fe + base;
  float Stm = 0.f, S1 = 0.f, S2 = 0.f, So = 0.f, F = 0.f;
  for (int k = 0; k < 64; ++k) {
    float o  = op[k * 512];
    float tm = tp[k * 512];
    float f  = fp[k * 512];
    float mo = o * tm;
    Stm += tm;
    S1  += mo;
    S2  += mo * mo;
    So  += o;
    F   += f * w[128 + k];
  }
  float denom = Stm + EPS;
  float mean  = S1 / denom;
  float var   = (S2 - 2.f * mean * S1 + 64.f * mean * mean) / denom;
  float stdv  = sqrtf(fmaxf(var, 0.f));
  float m     = So * (1.f / 64.f);

  // ---- T(b,l) = <TE[b,l,:], wt> via V_WMMA_F32_16X16X4_F32 ----
  // A (16x4): wt chunk broadcast across all M rows.
  //   lanes 0-15 hold {K=0 (v0), K=1 (v1)}, lanes 16-31 hold {K=2, K=3}.
  // B (4x16): B[k][n] = TE[l_row+n][i0+k], mirrored lane-half K split.
  // Since A is uniform over M, D is uniform over M: every lane's acc[0]
  // holds the dot product for its own N column => no shuffle needed.
  const bool hi   = lane >= 16;
  const int  lh   = lane & 15;
  const int  lrow = l0 + wave * 32;
  const float* te0 = te + ((size_t)b * 512 + lrow + lh) * 128;  // rows n=0..15
  const float* te1 = te0 + 16 * 128;                            // rows n=16..31
  v8f acc0 = {};
  v8f acc1 = {};
  for (int i0 = 0; i0 < 128; i0 += 4) {
    v2f a;
    a.x = w[i0 + (hi ? 2 : 0)];
    a.y = w[i0 + (hi ? 3 : 1)];
    float4 t0 = *(const float4*)(te0 + i0);
    float4 t1 = *(const float4*)(te1 + i0);
    v2f b0, b1;
    b0.x = hi ? t0.z : t0.x;
    b0.y = hi ? t0.w : t0.y;
    b1.x = hi ? t1.z : t1.x;
    b1.y = hi ? t1.w : t1.y;
    acc0 = __builtin_amdgcn_wmma_f32_16x16x4_f32(false, a, false, b0,
                                                 (short)0, acc0, false, false);
    acc1 = __builtin_amdgcn_wmma_f32_16x16x4_f32(false, a, false, b1,
                                                 (short)0, acc1, false, false);
  }
  float T = hi ? acc1[0] : acc0[0];

  float valp = T + F + w[192] * mean + w[193] * stdv + w[195];
  size_t idx = (size_t)b * 512 + l;
  m_out[idx]    = m;
  valp_out[idx] = valp;
}

// ---------------------------------------------------------------------------
// val = val_p + ws2 * resmean, where resmean uses the 5-wide rolling mean of
// the column mean (zero-padded 2 at each edge). 8192 threads.
// ---------------------------------------------------------------------------
__global__ __launch_bounds__(256) void finalize_kernel(
    const float* __restrict__ m, const float* __restrict__ valp,
    const float* __restrict__ w, float* __restrict__ val) {
  int idx = blockIdx.x * 256 + threadIdx.x;  // 0..8191
  int l = idx & 511;
  const float* mrow = m + (idx & ~511);
  float rm = 0.f;
  if (l >= 2 && l < 510) {
    float s = mrow[l - 2] + mrow[l - 1] + mrow[l] + mrow[l + 1] + mrow[l + 2];
    rm = mrow[l] - s * 0.2f;
  }
  val[idx] = valp[idx] + w[194] * rm;
}

// ---------------------------------------------------------------------------
// Broadcast val[B,L] -> out[B,32,64,L]. 64 MB of float4 stores: the only
// bandwidth-relevant kernel (~3 us floor at 23.3 TB/s). Each thread owns one
// (b,k,l4) float4 and stores the 32 channel replicas.
// ---------------------------------------------------------------------------
__global__ __launch_bounds__(256) void bcast_kernel(
    const float* __restrict__ val, float* __restrict__ out) {
  int tid = blockIdx.x * 256 + threadIdx.x;  // 0..131071
  int l4 = tid & 127;          // 128 float4 per row of 512
  int k  = (tid >> 7) & 63;
  int b  = tid >> 13;
  float4 v = ((const float4*)(val + (size_t)b * 512))[l4];
  float4* o = (float4*)out;
  size_t base = ((size_t)b * 32 * 64 + k) * 128 + l4;
#pragma unroll 4
  for (int c = 0; c < 32; ++c) {
    o[base + (size_t)c * 64 * 128] = v;
  }
}

// ---------------------------------------------------------------------------
extern "C" void kernel_launch(void* const* d_in, const int* in_sizes, int n_in,
                              void* d_out, int out_size, void* d_ws,
                              size_t ws_size, hipStream_t stream) {
  const float* obs     = (const float*)d_in[0];   // observed_data [16,64,512]
  // d_in[1] x_mask, d_in[2] cond_mask: dead in the reference output
  const float* tmask   = (const float*)d_in[3];   // target_mask  [16,64,512]
  const float* te      = (const float*)d_in[4];   // time_embed   [16,512,128]
  const float* fe      = (const float*)d_in[5];   // feature_embed[16,64,512]
  const float* W_stat  = (const float*)d_in[6];   // [128,3]
  const float* b_stat  = (const float*)d_in[7];   // [128]
  const float* W_time  = (const float*)d_in[8];   // [128,128]
  const float* b_time  = (const float*)d_in[9];   // [128]
  const float* W_feat  = (const float*)d_in[10];  // [16,64]
  const float* b_feat  = (const float*)d_in[11];  // [16]
  const float* W_final = (const float*)d_in[12];  // [1,272]
  const float* b_final = (const float*)d_in[13];  // [1]

  float* w    = (float*)d_ws;
  float* m    = w + 256;
  float* valp = w + 256 + 8192;
  float* val  = w + 256 + 16384;
  float* out  = (float*)d_out;

  prep_kernel<<<1, 256, 0, stream>>>(W_stat, b_stat, W_time, b_time, W_feat,
                                     b_feat, W_final, b_final, w);
  stats_wmma_kernel<<<32, 256, 0, stream>>>(obs, tmask, te, fe, w, m, valp);
  finalize_kernel<<<32, 256, 0, stream>>>(m, valp, w, val);
  bcast_kernel<<<512, 256, 0, stream>>>(val, out);
}